// TensorProduct_18829136626101
// MI455X (gfx1250) — compile-verified
//
#include <hip/hip_runtime.h>
#include <hip/hip_bf16.h>
#include <stdint.h>

// ---------------------------------------------------------------------------
// e3nn-style uvu tensor product, Z=200k, MUL=64, 5 paths.
// Memory-bound (~0.5 flop/byte, ~0.82 GB total traffic -> ~35us at 23.3 TB/s).
// CDNA5 path: Tensor Data Mover stages the two big streaming operands
// (x1 [Z,256], weight [Z,320]) into LDS (TENSORcnt-tracked, no VGPRs),
// VALU does the per-(z,u) irrep math, stores are coalesced.
// x2 [Z,4] is wave-uniform per 64-lane z-group -> scalar (SMEM) load.
// ---------------------------------------------------------------------------

#define ZT  16   // z rows per block
#define ZPW 2    // z rows DMA'd by each of the 8 waves

typedef __attribute__((ext_vector_type(4))) unsigned int v4u;
typedef __attribute__((ext_vector_type(8))) int          v8i;
typedef __attribute__((ext_vector_type(4))) int          v4i;

// Build a 1-D TDM descriptor (D#) and issue TENSOR_LOAD_TO_LDS.
// Layout per CDNA5 ISA:
//   group0: [1:0]=count=1, [63:32]=lds_addr, [120:64]=global_addr, [127:126]=type=2
//   group1: [17:16]=data_size(2 -> 4B), tensor_dim0=n, tensor_dim1=1,
//           tile_dim0=n, tile_dim1=1, tile_dim2=0 (unused), dim0_stride=n
__device__ __forceinline__ void tdm_load_1d(const float* gsrc,
                                            uint32_t lds_byte_off,
                                            uint32_t nfloat) {
  uint64_t ga = (uint64_t)(uintptr_t)gsrc;
  v4u g0;
  g0.x = 1u;                                                  // count=1 (valid)
  g0.y = lds_byte_off;                                        // LDS byte address
  g0.z = (uint32_t)ga;                                        // global_addr[31:0]
  g0.w = ((uint32_t)(ga >> 32) & 0x01FFFFFFu) | (2u << 30);   // [56:32] | type=2

  uint32_t n = nfloat;                                        // n <= 640 here
  v8i g1;
  g1[0] = (int)(2u << 16);                                    // data_size = 4B
  g1[1] = (int)((n & 0xFFFFu) << 16);                         // tensor_dim0 lo16
  g1[2] = (int)(((n >> 16) & 0xFFFFu) | (1u << 16));          // tdim0 hi16 | tdim1 lo16=1
  g1[3] = (int)((n & 0xFFFFu) << 16);                         // tdim1 hi16=0 | tile_dim0=n
  g1[4] = (int)1;                                             // tile_dim1=1, tile_dim2=0
  g1[5] = (int)n;                                             // tensor_dim0_stride lo32
  g1[6] = 0;                                                  // stride hi | dim1_stride lo
  g1[7] = 0;                                                  // dim1_stride hi
  v4i gz4 = {0, 0, 0, 0};                                     // groups 2/3 unused (1-D)
  v8i gz8 = {0, 0, 0, 0, 0, 0, 0, 0};
  __builtin_amdgcn_tensor_load_to_lds(g0, g1, gz4, gz4, gz8, 0);
}

__global__ __launch_bounds__(256)
void tp_uvu_kernel(const float* __restrict__ x1, const float* __restrict__ x2,
                   const float* __restrict__ w, float* __restrict__ out, int Z) {
  __shared__ float sx1[ZT * 256];
  __shared__ float sw [ZT * 320];

  const int z0   = blockIdx.x * ZT;
  const int zcnt = min(ZT, Z - z0);

  // Each wave DMAs its own contiguous 2-z slice of x1 and weight into LDS.
  const int wid = __builtin_amdgcn_readfirstlane((int)(threadIdx.x >> 5));
  const int zs  = wid * ZPW;
  const int zn  = min(ZPW, zcnt - zs);
  if (zn > 0) {
    uint32_t sx1_off = (uint32_t)(uintptr_t)(&sx1[zs * 256]);
    uint32_t sw_off  = (uint32_t)(uintptr_t)(&sw [zs * 320]);
    tdm_load_1d(x1 + (size_t)(z0 + zs) * 256, sx1_off, (uint32_t)(zn * 256));
    tdm_load_1d(w  + (size_t)(z0 + zs) * 320, sw_off,  (uint32_t)(zn * 320));
    __builtin_amdgcn_s_wait_tensorcnt(0);   // own wave's DMAs complete
  }
  __syncthreads();                          // publish LDS to all waves

  const float K0   = 0.70710678118654752f;  // 1/sqrt(2)
  const float K110 = 0.40824829046386302f;  // 1/sqrt(6)

#pragma unroll
  for (int it = 0; it < (ZT * 64) / 256; ++it) {
    int idx = (int)threadIdx.x + it * 256;  // (z_local, u) pair
    int zl  = idx >> 6;                     // wave-uniform (wave32: idx>>6 same
    int u   = idx & 63;                     //  for all 32 lanes of a wave)
    if (zl >= zcnt) break;

    // x2 row for this z: uniform across the wave -> scalar load path.
    int zlu = __builtin_amdgcn_readfirstlane(zl);
    const float* B = x2 + (size_t)(z0 + zlu) * 4;
    float b0 = B[0], bx = B[1], by = B[2], bz = B[3];

    const float* X = &sx1[zl * 256];
    const float* W = &sw [zl * 320];

    float a0 = X[u];
    float ax = X[64 + 3 * u + 0];
    float ay = X[64 + 3 * u + 1];
    float az = X[64 + 3 * u + 2];
    float w0 = W[u];
    float w1 = W[64  + u];
    float w2 = W[128 + u];
    float w3 = W[192 + u];
    float w4 = W[256 + u];

    float dot = ax * bx + ay * by + az * bz;
    float o0  = K0 * w0 * a0 * b0 + K110 * w3 * dot;          // 64x0e

    float o1x = K0 * (w1 * ax * b0 + w2 * a0 * bx);           // 64x1o
    float o1y = K0 * (w1 * ay * b0 + w2 * a0 * by);
    float o1z = K0 * (w1 * az * b0 + w2 * a0 * bz);

    float cxp = ay * bz - az * by;                            // a1 x b1
    float cyp = az * bx - ax * bz;
    float czp = ax * by - ay * bx;
    float o2x = K0 * w4 * cxp;                                // 64x1e
    float o2y = K0 * w4 * cyp;
    float o2z = K0 * w4 * czp;

    float* O = out + (size_t)(z0 + zl) * 448;
    O[u] = o0;
    O[64  + 3 * u + 0] = o1x;
    O[64  + 3 * u + 1] = o1y;
    O[64  + 3 * u + 2] = o1z;
    O[256 + 3 * u + 0] = o2x;
    O[256 + 3 * u + 1] = o2y;
    O[256 + 3 * u + 2] = o2z;
  }
}

extern "C" void kernel_launch(void* const* d_in, const int* in_sizes, int n_in,
                              void* d_out, int out_size, void* d_ws, size_t ws_size,
                              hipStream_t stream) {
  const float* x1 = (const float*)d_in[0];   // [Z, 256]
  const float* x2 = (const float*)d_in[1];   // [Z, 4]
  const float* w  = (const float*)d_in[2];   // [Z, 320]
  float*       out = (float*)d_out;          // [Z, 448]

  const int Z = in_sizes[0] / 256;
  const int blocks = (Z + ZT - 1) / ZT;
  tp_uvu_kernel<<<blocks, 256, 0, stream>>>(x1, x2, w, out, Z);
}